// TemporalAttention_54820962566762
// MI455X (gfx1250) — compile-verified
//
#include <hip/hip_runtime.h>
#include <hip/hip_bf16.h>

// TemporalAttention for MI455X (gfx1250, wave32).
// H: (B=16, T=64, C=64, H=32, W=32) fp32 = 256 MB. Memory-bound (AI ~0.5 F/B):
// floor = ~516 MB / 23.3 TB/s ~= 22 us. Strategy: two full streaming passes over
// H; dot-product pass runs through V_WMMA_F32_16X16X4_F32 (free compute on the
// matrix pipe), weighted-sum pass is pure coalesced b128 streaming.

typedef float v2f __attribute__((ext_vector_type(2)));
typedef float v8f __attribute__((ext_vector_type(8)));

#define BATCH 16
#define TSTEPS 64
#define DK 65536            // C*H*W = 64*32*32
#define KSLICES 32          // k-split for score partials
#define KS_LEN (DK / KSLICES)        // 2048 k per workgroup
#define KW_LEN (KS_LEN / 8)          // 256 k per wave -> 64 WMMA k-steps

// -------- Pass 1: partial scores via V_WMMA_F32_16X16X4_F32 ------------------
// grid.x = BATCH * 4(t-tiles) * KSLICES = 2048, block = 256 (8 waves).
// A[m][k] = last[b, k]  (rows broadcast; documented fp32 A 16x4 layout:
//   VGPR0 = K0 (lanes 0-15) / K2 (lanes 16-31), VGPR1 = K1 / K3)
// B[k][n] = H[b, t0+n, k] (lane n holds column n, same K-per-VGPR split)
// => D[m][n] = sum_k last[k] * H[t0+n, k]  (all rows m identical)
__global__ __launch_bounds__(256) void ta_scores_partial(
    const float* __restrict__ Hs, float* __restrict__ partial)
{
    const int wg = blockIdx.x;
    const int ks = wg & (KSLICES - 1);
    const int tt = (wg >> 5) & 3;            // t-tile (16 t's each)
    const int b  = wg >> 7;

    const int lane = threadIdx.x & 31;
    const int wave = threadIdx.x >> 5;
    const int n  = lane & 15;                // column = t within tile
    const int hi = lane >> 4;                // half-wave selects K pair 0/1 vs 2/3

    const float* __restrict__ vlast =
        Hs + ((size_t)b * TSTEPS + (TSTEPS - 1)) * DK;
    const float* __restrict__ hrow =
        Hs + ((size_t)b * TSTEPS + (tt * 16 + n)) * DK;

    const int k0 = ks * KS_LEN + wave * KW_LEN;
    const float* ap = vlast + k0 + 2 * hi;   // A: broadcast last-step vector
    const float* bp = hrow  + k0 + 2 * hi;   // B: this lane's t-row (strided)

    v8f c = {0.f, 0.f, 0.f, 0.f, 0.f, 0.f, 0.f, 0.f};
    for (int o = 0; o < KW_LEN / (4 * 16); ++o) {      // 4 outer chunks
        __builtin_prefetch(bp + 64, 0, 0);             // global_prefetch_b8, 256B ahead
        #pragma unroll
        for (int i = 0; i < 16; ++i) {                 // 16 WMMAs, k advances by 4
            v2f a  = *reinterpret_cast<const v2f*>(ap);
            v2f bb = *reinterpret_cast<const v2f*>(bp);
            c = __builtin_amdgcn_wmma_f32_16x16x4_f32(
                    /*neg_a=*/false, a, /*neg_b=*/false, bb,
                    /*c_mod=*/(short)0, c, /*reuse_a=*/false, /*reuse_b=*/false);
            ap += 4; bp += 4;
        }
    }

    // All rows of D identical -> c[0] on lanes 0-15 = partial score for t0+n.
    __shared__ float red[8][16];
    if (lane < 16) red[wave][n] = c[0];
    __syncthreads();
    if (threadIdx.x < 16) {
        float s = 0.f;
        #pragma unroll
        for (int w = 0; w < 8; ++w) s += red[w][threadIdx.x];
        partial[(size_t)ks * (BATCH * TSTEPS) + b * TSTEPS + tt * 16 + threadIdx.x] = s;
    }
}

// -------- Pass 2: deterministic k-slice reduction + softmax ------------------
// grid.x = BATCH, block = 64 (one thread per t).
__global__ __launch_bounds__(64) void ta_softmax(
    const float* __restrict__ partial, float* __restrict__ weights)
{
    const int b = blockIdx.x;
    const int t = threadIdx.x;

    float s = 0.f;
    #pragma unroll
    for (int ks = 0; ks < KSLICES; ++ks)
        s += partial[(size_t)ks * (BATCH * TSTEPS) + b * TSTEPS + t];
    s *= (1.0f / (float)TSTEPS);

    __shared__ float sh[TSTEPS];
    __shared__ float smax, ssum;
    sh[t] = s;
    __syncthreads();
    if (t == 0) {
        float m = sh[0];
        for (int i = 1; i < TSTEPS; ++i) m = fmaxf(m, sh[i]);
        smax = m;
    }
    __syncthreads();
    const float e = expf(s - smax);
    sh[t] = e;
    __syncthreads();
    if (t == 0) {
        float su = 0.f;
        for (int i = 0; i < TSTEPS; ++i) su += sh[i];
        ssum = su;
    }
    __syncthreads();
    weights[b * TSTEPS + t] = e / ssum;
}

// -------- Pass 3: context = sum_t w[t] * H[t, :] (pure b128 streaming) -------
// grid.x = BATCH * 64 = 1024, block = 256; one float4 output per thread.
__global__ __launch_bounds__(256) void ta_context(
    const float* __restrict__ Hs, const float* __restrict__ weights,
    float* __restrict__ out)
{
    const int b     = blockIdx.x >> 6;
    const int chunk = blockIdx.x & 63;           // 64 chunks of 1024 floats
    const int idx   = (chunk * 1024 + threadIdx.x * 4) >> 2;  // float4 index

    __shared__ float w[TSTEPS];
    if (threadIdx.x < TSTEPS) w[threadIdx.x] = weights[b * TSTEPS + threadIdx.x];
    __syncthreads();

    const float4* __restrict__ Hb =
        reinterpret_cast<const float4*>(Hs + (size_t)b * TSTEPS * DK);
    float4 acc = {0.f, 0.f, 0.f, 0.f};
    #pragma unroll 8
    for (int t = 0; t < TSTEPS; ++t) {
        const float4 h = Hb[(size_t)t * (DK / 4) + idx];
        const float wt = w[t];
        acc.x += h.x * wt; acc.y += h.y * wt;
        acc.z += h.z * wt; acc.w += h.w * wt;
    }
    reinterpret_cast<float4*>(out)[(size_t)b * (DK / 4) + idx] = acc;
}

extern "C" void kernel_launch(void* const* d_in, const int* in_sizes, int n_in,
                              void* d_out, int out_size, void* d_ws, size_t ws_size,
                              hipStream_t stream) {
    (void)in_sizes; (void)n_in; (void)out_size; (void)ws_size;
    const float* Hs = (const float*)d_in[0];
    float* out = (float*)d_out;

    // ws layout: [0, 32768) partial scores (KSLICES * B * T floats),
    //            [32768, 33792) softmax weights (B * T floats).  ~135 KB total.
    float* partial = (float*)d_ws;
    float* weights = partial + KSLICES * BATCH * TSTEPS;

    ta_scores_partial<<<BATCH * 4 * KSLICES, 256, 0, stream>>>(Hs, partial);
    ta_softmax<<<BATCH, 64, 0, stream>>>(partial, weights);
    ta_context<<<BATCH * 64, 256, 0, stream>>>(Hs, weights, out);
}